// CAGAM3D_88098369176270
// MI455X (gfx1250) — compile-verified
//
#include <hip/hip_runtime.h>
#include <hip/hip_bf16.h>
#include <math.h>

typedef float v2f __attribute__((ext_vector_type(2)));
typedef float v4f __attribute__((ext_vector_type(4)));
typedef float v8f __attribute__((ext_vector_type(8)));

#define BATCH 2
#define CIN   512
#define NI    6
#define NC    10
#define NSP   4096          // 16*16*16
#define LDSW  516           // LDS row stride (dwords): %64==4 -> conflict-free A reads

// ---------------------------------------------------------------------------
// Kernel 1: context = relu(Wc @ features + bc); then Q,K,V = 6x6 pconvs.
// One wave (32 threads) per (batch, 16-column tile). WMMA f32 16x16x4,
// M = out-channels (6 valid of 16), N = 16 spatial columns, K = 512.
// Two independent WMMA accumulation chains hide the D->C dependency.
// ---------------------------------------------------------------------------
__global__ __launch_bounds__(32) void ctx_qkv_wmma(
    const float* __restrict__ feat,
    const float* __restrict__ Wc, const float* __restrict__ bc,
    const float* __restrict__ Wq, const float* __restrict__ bq,
    const float* __restrict__ Wk, const float* __restrict__ bk,
    const float* __restrict__ Wv, const float* __restrict__ bv,
    float* __restrict__ ctx, float* __restrict__ qf,
    float* __restrict__ kf,  float* __restrict__ vf)
{
    __shared__ float sW[16 * LDSW];
    const int lane = threadIdx.x;      // 0..31
    const int half = lane >> 4;        // 0 or 1
    const int l16  = lane & 15;

    // Rows 0..5: copy Wc with float4 loads/stores (LDSW%4==0 -> b128 stores).
    // 6*512 floats = 768 float4 -> 24 iterations per lane.
    {
        const v4f* wc4 = (const v4f*)Wc;
        for (int i4 = lane; i4 < NI * (CIN / 4); i4 += 32) {
            int o  = i4 >> 7;              // /128 float4 per row
            int cc = (i4 & 127) << 2;
            *(v4f*)&sW[o * LDSW + cc] = wc4[i4];
        }
        // Rows 6..15: zero (10*512 floats = 1280 float4).
        v4f z = {0.f, 0.f, 0.f, 0.f};
        for (int i4 = lane; i4 < 10 * (CIN / 4); i4 += 32) {
            int o  = NI + (i4 >> 7);
            int cc = (i4 & 127) << 2;
            *(v4f*)&sW[o * LDSW + cc] = z;
        }
    }
    __syncthreads();

    const int blk = blockIdx.x;        // 0..511
    const int b   = blk >> 8;
    const int n0  = (blk & 255) << 4;
    const int col = n0 + l16;
    const float* fb = feat + (size_t)b * CIN * NSP;

    v8f acc0 = {0.f, 0.f, 0.f, 0.f, 0.f, 0.f, 0.f, 0.f};
    v8f acc1 = {0.f, 0.f, 0.f, 0.f, 0.f, 0.f, 0.f, 0.f};

    // A 16x4 layout: lanes 0-15 -> M=lane, VGPR0/1 = K{0,1}; lanes 16-31 = K{2,3}
    // B 4x16 layout (symmetric with C/D): VGPR0 = K0 (lanes 0-15) / K2 (16-31)
    for (int k = 0; k < CIN; k += 8) {
        const int ka = k + 2 * half;
        v2f a0, b0, a1, b1;
        a0.x = sW[l16 * LDSW + ka];
        a0.y = sW[l16 * LDSW + ka + 1];
        a1.x = sW[l16 * LDSW + ka + 4];
        a1.y = sW[l16 * LDSW + ka + 5];
        const float* fp = fb + (size_t)ka * NSP + col;
        b0.x = fp[0];
        b0.y = fp[NSP];
        b1.x = fp[4 * NSP];
        b1.y = fp[5 * NSP];
        acc0 = __builtin_amdgcn_wmma_f32_16x16x4_f32(
            false, a0, false, b0, (short)0, acc0, false, false);
        acc1 = __builtin_amdgcn_wmma_f32_16x16x4_f32(
            false, a1, false, b1, (short)0, acc1, false, false);
    }
    v8f acc = acc0 + acc1;

    // C/D layout: lane<16 holds rows M=0..7 (all NI=6 valid rows) of column col.
    if (half == 0) {
        float cv[NI];
#pragma unroll
        for (int r = 0; r < NI; ++r) {
            float v = acc[r] + bc[r];
            cv[r] = v > 0.0f ? v : 0.0f;
            ctx[((size_t)b * NI + r) * NSP + col] = cv[r];
        }
#pragma unroll
        for (int o = 0; o < NI; ++o) {
            float aq = bq[o], ak = bk[o], av = bv[o];
#pragma unroll
            for (int c = 0; c < NI; ++c) {
                aq += Wq[o * NI + c] * cv[c];
                ak += Wk[o * NI + c] * cv[c];
                av += Wv[o * NI + c] * cv[c];
            }
            qf[((size_t)b * NI + o) * NSP + col] = aq;
            kf[((size_t)b * NI + o) * NSP + col] = ak;
            vf[((size_t)b * NI + o) * NSP + col] = av;
        }
    }
}

// ---------------------------------------------------------------------------
// Kernel 2: QD = Wcq@cam + bcq ; KD = Wck@cam + bck   (6x6 pconvs)
// ---------------------------------------------------------------------------
__global__ __launch_bounds__(256) void cam_qdkd(
    const float* __restrict__ cam,
    const float* __restrict__ Wcq, const float* __restrict__ bcq,
    const float* __restrict__ Wck, const float* __restrict__ bck,
    float* __restrict__ qdf, float* __restrict__ kdf)
{
    int i = blockIdx.x * blockDim.x + threadIdx.x;
    if (i >= BATCH * NSP) return;
    int b = i >> 12, n = i & (NSP - 1);
    float c[NI];
#pragma unroll
    for (int d = 0; d < NI; ++d) c[d] = cam[((size_t)b * NI + d) * NSP + n];
#pragma unroll
    for (int o = 0; o < NI; ++o) {
        float aq = bcq[o], ak = bck[o];
#pragma unroll
        for (int d = 0; d < NI; ++d) {
            aq += Wcq[o * NI + d] * c[d];
            ak += Wck[o * NI + d] * c[d];
        }
        qdf[((size_t)b * NI + o) * NSP + n] = aq;
        kdf[((size_t)b * NI + o) * NSP + n] = ak;
    }
}

// ---------------------------------------------------------------------------
// Kernel 3: fused dual-low-rank scores + online softmax + V-accumulate +
// residual + output pconv. ONE WAVE32 PER ROW n: lane L handles columns
// m = L, L+32, ... (coalesced K/KD/V loads), then the 32 partial online-
// softmax states are merged with a 5-step shfl_xor log-sum-exp butterfly.
// Never materializes the [N,N] attention matrix.
// ---------------------------------------------------------------------------
__global__ __launch_bounds__(256) void attn_out(
    const float* __restrict__ ctx, const float* __restrict__ qf,
    const float* __restrict__ kf,  const float* __restrict__ vf,
    const float* __restrict__ qdf, const float* __restrict__ kdf,
    const float* __restrict__ Wo,  const float* __restrict__ bo,
    const float* __restrict__ beta, float* __restrict__ out)
{
    const int lane = threadIdx.x & 31;
    const int row  = blockIdx.x * (blockDim.x >> 5) + (threadIdx.x >> 5);
    if (row >= BATCH * NSP) return;
    const int b = row >> 12, n = row & (NSP - 1);
    const float invscale = 0.40824829046386307f;   // 1/sqrt(6)

    float q[NI], qd[NI], acc[NI];
#pragma unroll
    for (int d = 0; d < NI; ++d) {
        q[d]  = qf[((size_t)b * NI + d) * NSP + n];   // row-uniform broadcast
        qd[d] = qdf[((size_t)b * NI + d) * NSP + n];
        acc[d] = 0.0f;
    }

    const float* kb  = kf  + (size_t)b * NI * NSP;
    const float* kdb = kdf + (size_t)b * NI * NSP;
    const float* vb  = vf  + (size_t)b * NI * NSP;

    float Mx = -INFINITY, L = 0.0f;
    for (int m = lane; m < NSP; m += 32) {
        float s1 = 0.0f, s2 = 0.0f;
#pragma unroll
        for (int d = 0; d < NI; ++d) {
            s1 += q[d]  * kb[d * NSP + m];
            s2 += qd[d] * kdb[d * NSP + m];
        }
        float s    = s1 * s2 * invscale;
        float newM = fmaxf(Mx, s);
        float corr = __expf(Mx - newM);
        float p    = __expf(s - newM);
        L = L * corr + p;
#pragma unroll
        for (int d = 0; d < NI; ++d)
            acc[d] = acc[d] * corr + p * vb[d * NSP + m];
        Mx = newM;
    }

    // Butterfly merge of 32 online-softmax partial states.
#pragma unroll
    for (int off = 16; off >= 1; off >>= 1) {
        float oM = __shfl_xor(Mx, off, 32);
        float oL = __shfl_xor(L,  off, 32);
        float oA[NI];
#pragma unroll
        for (int d = 0; d < NI; ++d) oA[d] = __shfl_xor(acc[d], off, 32);
        float newM = fmaxf(Mx, oM);
        float s1 = __expf(Mx - newM);
        float s2 = __expf(oM - newM);
        L = L * s1 + oL * s2;
#pragma unroll
        for (int d = 0; d < NI; ++d) acc[d] = acc[d] * s1 + oA[d] * s2;
        Mx = newM;
    }

    const float beta0 = beta[0];
    const float invL  = 1.0f / L;
    float enh[NI];
#pragma unroll
    for (int d = 0; d < NI; ++d)
        enh[d] = ctx[((size_t)b * NI + d) * NSP + n] + beta0 * (acc[d] * invL);

    // Lanes 0..9 each emit one output channel.
    if (lane < NC) {
        const int o = lane;
        float v = bo[o];
#pragma unroll
        for (int d = 0; d < NI; ++d) v += Wo[o * NI + d] * enh[d];
        out[((size_t)b * NC + o) * NSP + n] = v;
    }
}

// ---------------------------------------------------------------------------
// Kernel 4: logits[b,o] = mean over N of output_maps. One block per (b,o).
// ---------------------------------------------------------------------------
__global__ __launch_bounds__(256) void logits_reduce(
    const float* __restrict__ out_maps, float* __restrict__ logits)
{
    __shared__ float red[256];
    const int bo = blockIdx.x;             // 0..19
    const float* p = out_maps + (size_t)bo * NSP;
    float s = 0.0f;
    for (int i = threadIdx.x; i < NSP; i += 256) s += p[i];
    red[threadIdx.x] = s;
    __syncthreads();
    for (int off = 128; off > 0; off >>= 1) {
        if (threadIdx.x < off) red[threadIdx.x] += red[threadIdx.x + off];
        __syncthreads();
    }
    if (threadIdx.x == 0) logits[bo] = red[0] * (1.0f / NSP);
}

// ---------------------------------------------------------------------------
extern "C" void kernel_launch(void* const* d_in, const int* in_sizes, int n_in,
                              void* d_out, int out_size, void* d_ws, size_t ws_size,
                              hipStream_t stream) {
    const float* feat = (const float*)d_in[0];
    const float* cam  = (const float*)d_in[1];
    const float* Wc   = (const float*)d_in[2];
    const float* bc   = (const float*)d_in[3];
    const float* Wq   = (const float*)d_in[4];
    const float* bq   = (const float*)d_in[5];
    const float* Wk   = (const float*)d_in[6];
    const float* bk   = (const float*)d_in[7];
    const float* Wv   = (const float*)d_in[8];
    const float* bv   = (const float*)d_in[9];
    const float* Wcq  = (const float*)d_in[10];
    const float* bcq  = (const float*)d_in[11];
    const float* Wck  = (const float*)d_in[12];
    const float* bck  = (const float*)d_in[13];
    const float* Wo   = (const float*)d_in[14];
    const float* bo   = (const float*)d_in[15];
    const float* beta = (const float*)d_in[16];

    float* ws = (float*)d_ws;
    const size_t arr = (size_t)BATCH * NI * NSP;   // 49152 floats
    float* ctx = ws;
    float* qf  = ws + 1 * arr;
    float* kf  = ws + 2 * arr;
    float* vf  = ws + 3 * arr;
    float* qdf = ws + 4 * arr;
    float* kdf = ws + 5 * arr;

    float* out_maps = (float*)d_out;                             // [2,10,16,16,16]
    float* logits   = (float*)d_out + (size_t)BATCH * NC * NSP;  // [2,10]

    ctx_qkv_wmma<<<BATCH * (NSP / 16), 32, 0, stream>>>(
        feat, Wc, bc, Wq, bq, Wk, bk, Wv, bv, ctx, qf, kf, vf);

    cam_qdkd<<<(BATCH * NSP + 255) / 256, 256, 0, stream>>>(
        cam, Wcq, bcq, Wck, bck, qdf, kdf);

    // one wave32 per row: 8 rows per 256-thread block -> 1024 blocks
    attn_out<<<(BATCH * NSP) / 8, 256, 0, stream>>>(
        ctx, qf, kf, vf, qdf, kdf, Wo, bo, beta, out_maps);

    logits_reduce<<<BATCH * NC, 256, 0, stream>>>(out_maps, logits);
}